// GNNQLearning_54898271977543
// MI455X (gfx1250) — compile-verified
//
#include <hip/hip_runtime.h>
#include <hip/hip_bf16.h>

// GCN (1->32->16->1) on MI455X / gfx1250.
// Edge propagation is L2-resident (all node buffers < 40MB << 192MB L2);
// HBM traffic is dominated by the int64 edge-list streams.
// Layer-2 GEMM [N,32]@[32,16] uses V_WMMA_F32_16X16X4_F32 (full f32 precision),
// 8 chained K=4 WMMAs per 16x16 output tile, one wave per tile.

typedef __attribute__((ext_vector_type(2))) float v2f;
typedef __attribute__((ext_vector_type(8))) float v8f;

#define TPB 256

__global__ __launch_bounds__(TPB) void k_fill1(float* __restrict__ p, int n) {
  int t = blockIdx.x * TPB + threadIdx.x;
  if (t < n) p[t] = 1.0f;  // self-loop contributes 1 to every node's degree
}

__global__ __launch_bounds__(TPB) void k_deg(const long long* __restrict__ dst,
                                             float* __restrict__ deg, int E) {
  int t = blockIdx.x * TPB + threadIdx.x;
  if (t < E) unsafeAtomicAdd(&deg[(int)dst[t]], 1.0f);
}

__global__ __launch_bounds__(TPB) void k_rsqrt(float* __restrict__ p, int n) {
  int t = blockIdx.x * TPB + threadIdx.x;
  if (t < n) p[t] = rsqrtf(p[t]);  // deg >= 1 always (self-loops), no where() needed
}

// h1[n][f] = x[n] * W1[f]   (Fin == 1)
__global__ __launch_bounds__(TPB) void k_lin1(const float* __restrict__ x,
                                              const float* __restrict__ W1,
                                              float* __restrict__ h1, int total) {
  int t = blockIdx.x * TPB + threadIdx.x;
  if (t < total) {
    int n = t >> 5, f = t & 31;
    h1[t] = x[n] * W1[f];
  }
}

// out[n][f] = h[n][f] * dis[n]^2   (the self-loop edge, norm = dis[n]*dis[n])
template <int LOGF>
__global__ __launch_bounds__(TPB) void k_selfinit(const float* __restrict__ h,
                                                  const float* __restrict__ dis,
                                                  float* __restrict__ out, int total) {
  int t = blockIdx.x * TPB + threadIdx.x;
  if (t < total) {
    int n = t >> LOGF;
    float d = dis[n];
    out[t] = h[t] * d * d;
  }
}

// One thread per (edge, feature). A wave covers all F features of 32/F edges:
// src/dst/dis loads are wave-uniform per edge, h loads and atomics are coalesced.
template <int LOGF>
__global__ __launch_bounds__(TPB) void k_prop(const long long* __restrict__ src,
                                              const long long* __restrict__ dst,
                                              const float* __restrict__ dis,
                                              const float* __restrict__ h,
                                              float* __restrict__ out, int total) {
  int t = blockIdx.x * TPB + threadIdx.x;
  if (t >= total) return;
  int e = t >> LOGF;
  int f = t & ((1 << LOGF) - 1);
  int s = (int)src[e];
  int d = (int)dst[e];
  float w = dis[s] * dis[d];
  unsafeAtomicAdd(&out[((size_t)d << LOGF) + f], h[((size_t)s << LOGF) + f] * w);
}

// Layer-2 GEMM: h2[16x16 tile] = relu(out1 + b1) @ W2, via V_WMMA_F32_16X16X4_F32.
// One wave per 16-node tile. ISA 7.12.2 layouts:
//   A (16x4 f32):  lane L holds row M=L&15; VGPR0/1 = K = 4i + 2*(L>>4) + {0,1}
//   B (4x16 f32):  lane L holds col N=L&15; VGPR0/1 = K = 4i + 2*(L>>4) + {0,1}
//   C/D (16x16):   VGPR v, lane L -> row v + 8*(L>>4), col L&15
__global__ __launch_bounds__(TPB) void k_gemm2_wmma(const float* __restrict__ out1,
                                                    const float* __restrict__ b1,
                                                    const float* __restrict__ W2,
                                                    float* __restrict__ h2,
                                                    int n_tiles) {
  int wave = (blockIdx.x * TPB + threadIdx.x) >> 5;
  if (wave >= n_tiles) return;          // whole wave exits together: EXEC stays all-1 for WMMA
  int lane = threadIdx.x & 31;
  int row  = lane & 15;                 // A row (M) and B column (N)
  int hi   = lane >> 4;                 // 0: K%4 in {0,1}; 1: K%4 in {2,3}
  const float* arow = out1 + (size_t)(wave * 16 + row) * 32;

  v8f c = {};
#pragma unroll
  for (int i = 0; i < 8; ++i) {
    int k = 4 * i + 2 * hi;
    v2f a, b;
    a.x = fmaxf(arow[k]     + b1[k],     0.0f);   // bias + ReLU fused into A load
    a.y = fmaxf(arow[k + 1] + b1[k + 1], 0.0f);
    b.x = W2[(k)     * 16 + row];                 // W2 is row-major [32,16]
    b.y = W2[(k + 1) * 16 + row];
    c = __builtin_amdgcn_wmma_f32_16x16x4_f32(
        /*neg_a=*/false, a, /*neg_b=*/false, b,
        /*c_mod=*/(short)0, c, /*reuse_a=*/false, /*reuse_b=*/false);
  }

  size_t base = (size_t)wave * 16;
#pragma unroll
  for (int v = 0; v < 8; ++v) {
    int m = v + 8 * hi;
    h2[(base + m) * 16 + row] = c[v];
  }
}

// Scalar tail for N % 16 nodes (unused for N=100000 but keeps the kernel general).
__global__ __launch_bounds__(TPB) void k_gemm2_tail(const float* __restrict__ out1,
                                                    const float* __restrict__ b1,
                                                    const float* __restrict__ W2,
                                                    float* __restrict__ h2,
                                                    int n0, int total) {
  int t = blockIdx.x * TPB + threadIdx.x;
  if (t >= total) return;
  int n = n0 + (t >> 4);
  int j = t & 15;
  float acc = 0.0f;
  for (int k = 0; k < 32; ++k)
    acc += fmaxf(out1[(size_t)n * 32 + k] + b1[k], 0.0f) * W2[k * 16 + j];
  h2[(size_t)n * 16 + j] = acc;
}

// h3[n] = relu(out2[n] + b2) . W3   (16 -> 1)
__global__ __launch_bounds__(TPB) void k_lin3(const float* __restrict__ out2,
                                              const float* __restrict__ b2,
                                              const float* __restrict__ W3,
                                              float* __restrict__ h3, int n) {
  int t = blockIdx.x * TPB + threadIdx.x;
  if (t >= n) return;
  const float* r = out2 + (size_t)t * 16;
  float acc = 0.0f;
#pragma unroll
  for (int k = 0; k < 16; ++k)
    acc += fmaxf(r[k] + b2[k], 0.0f) * W3[k];
  h3[t] = acc;
}

// d_out[n] = h3[n]*dis[n]^2 + b3  (self-loop + bias; edge atomics accumulate on top)
__global__ __launch_bounds__(TPB) void k_final_init(const float* __restrict__ h3,
                                                    const float* __restrict__ dis,
                                                    const float* __restrict__ b3,
                                                    float* __restrict__ out, int n) {
  int t = blockIdx.x * TPB + threadIdx.x;
  if (t < n) {
    float d = dis[t];
    out[t] = h3[t] * d * d + b3[0];
  }
}

static inline int nblk(long long n) { return (int)((n + TPB - 1) / TPB); }

extern "C" void kernel_launch(void* const* d_in, const int* in_sizes, int n_in,
                              void* d_out, int out_size, void* d_ws, size_t ws_size,
                              hipStream_t stream) {
  const float*     x   = (const float*)d_in[0];
  const long long* ei  = (const long long*)d_in[1];   // int64 [2, E]
  const float*     W1  = (const float*)d_in[2];
  const float*     b1  = (const float*)d_in[3];
  const float*     W2  = (const float*)d_in[4];
  const float*     b2  = (const float*)d_in[5];
  const float*     W3  = (const float*)d_in[6];
  const float*     b3  = (const float*)d_in[7];
  float* out = (float*)d_out;

  const int N = in_sizes[0];       // x is [N, 1]
  const int E = in_sizes[1] / 2;
  const long long* src = ei;
  const long long* dst = ei + E;

  // Workspace layout (floats): dis[N] | bufA[N*32] | bufB[N*32]
  // bufA: h1 -> h2 -> h3 (each producer's input lives in bufB at that point)
  // bufB: out1 -> out2
  float* dis  = (float*)d_ws;
  float* bufA = dis + N;
  float* bufB = bufA + (size_t)N * 32;

  // --- degree / normalization ---
  k_fill1<<<nblk(N), TPB, 0, stream>>>(dis, N);
  k_deg  <<<nblk(E), TPB, 0, stream>>>(dst, dis, E);
  k_rsqrt<<<nblk(N), TPB, 0, stream>>>(dis, N);

  // --- layer 1: lin(1->32), self-loop init, edge scatter ---
  k_lin1<<<nblk((long long)N * 32), TPB, 0, stream>>>(x, W1, bufA, N * 32);
  k_selfinit<5><<<nblk((long long)N * 32), TPB, 0, stream>>>(bufA, dis, bufB, N * 32);
  k_prop<5><<<nblk((long long)E * 32), TPB, 0, stream>>>(src, dst, dis, bufA, bufB, E * 32);

  // --- layer 2: WMMA GEMM relu(out1+b1)@W2 -> h2 (into bufA), then propagate ---
  int tiles = N / 16;
  int rem   = N - tiles * 16;
  if (tiles > 0)
    k_gemm2_wmma<<<nblk((long long)tiles * 32), TPB, 0, stream>>>(bufB, b1, W2, bufA, tiles);
  if (rem > 0)
    k_gemm2_tail<<<nblk((long long)rem * 16), TPB, 0, stream>>>(bufB, b1, W2, bufA,
                                                                tiles * 16, rem * 16);
  k_selfinit<4><<<nblk((long long)N * 16), TPB, 0, stream>>>(bufA, dis, bufB, N * 16);
  k_prop<4><<<nblk((long long)E * 16), TPB, 0, stream>>>(src, dst, dis, bufA, bufB, E * 16);

  // --- layer 3: lin(16->1), final init (+b3), edge scatter into d_out ---
  k_lin3<<<nblk(N), TPB, 0, stream>>>(bufB, b2, W3, bufA, N);
  k_final_init<<<nblk(N), TPB, 0, stream>>>(bufA, dis, b3, out, N);
  k_prop<0><<<nblk(E), TPB, 0, stream>>>(src, dst, dis, bufA, out, E);
}